// HodgkinHuxleyODE_51153060495819
// MI455X (gfx1250) — compile-verified
//
#include <hip/hip_runtime.h>

// ---------------------------------------------------------------------------
// Hodgkin-Huxley elementwise derivative kernel for MI455X (gfx1250).
// Memory-bound streaming kernel: 40 B/element, ~335 MB total -> ~14.4 us floor
// at 23.3 TB/s. Strategy: 128-bit non-temporal global loads/stores, hardware
// v_exp_f32 / v_rcp_f32 transcendentals, branch-free selects, wave32 blocks.
// WMMA is mathematically inapplicable (no contraction dimension).
// ---------------------------------------------------------------------------

typedef __attribute__((ext_vector_type(4))) float v4f;

// I_in = CURR_LEVEL / A_SOMA, A_SOMA = pi * (70e-4)^2  (folded at compile time)
#define HH_I_CONST ((float)(0.0005 / (3.141592653589793 * 0.007 * 0.007)))

__device__ __forceinline__ float hh_rcp(float x) {
    return __builtin_amdgcn_rcpf(x);            // v_rcp_f32
}

// efun(z) = z/(exp(z)-1), Taylor 1 - z/2 for |z| < 1e-4 (branch-free select)
__device__ __forceinline__ float hh_efun(float z) {
    const float big   = z * hh_rcp(__expf(z) - 1.0f);   // v_exp_f32 + v_rcp_f32
    const float small = 1.0f - 0.5f * z;
    return (__builtin_fabsf(z) < 1e-4f) ? small : big;  // v_cmp + v_cndmask
}

__device__ __forceinline__ void hh_point(float V, float n, float m, float h, float p,
                                         float I_in,
                                         float& dV, float& dn, float& dm,
                                         float& dh, float& dp) {
    // Rate constants with VT = -60 folded into the offsets.
    const float a_m = 1.28f  * hh_efun(-0.25f * (V + 47.0f));
    const float b_m = 1.40f  * hh_efun( 0.20f * (V + 20.0f));
    const float a_h = 0.128f * __expf((V + 43.0f) * (-1.0f / 18.0f));
    const float b_h = 4.0f   * hh_rcp(1.0f + __expf(-0.2f * (V + 20.0f)));
    const float a_n = 0.16f  * hh_efun(-0.20f * (V + 45.0f));
    const float b_n = 0.5f   * __expf((V + 50.0f) * (-1.0f / 40.0f));

    // p gate: p_inf and 1/tau_p   (exp(-z) = rcp(exp(z)) saves one exp)
    const float v1      = V + 35.0f;
    const float p_inf   = hh_rcp(1.0f + __expf(-0.1f * v1));
    const float e_p     = __expf(0.05f * v1);
    const float inv_tau = (3.3f * e_p + hh_rcp(e_p)) * (1.0f / 600.0f);

    // Membrane equation (C_MEM = 1)
    const float m3  = m * m * m;
    const float n2  = n * n;
    const float ekv = -107.0f - V;                       // E_K - V
    dV = fmaf(m3 * 4.0f * h, 53.0f - V,                  // gNa * m^3 h (E_Na - V)
         fmaf(n2 * n2 * 1.5f, ekv,                       // gK  * n^4   (E_K  - V)
         fmaf(0.07f * p, ekv,                            // gM  * p     (E_K  - V)
         fmaf(0.1f, -70.0f - V, I_in))));                // gL (E_L - V) + I

    // (a/(a+b) - x)*(a+b) == a - (a+b)*x  -> single FMA each, no divide
    dn = fmaf(-(a_n + b_n), n, a_n);
    dm = fmaf(-(a_m + b_m), m, a_m);
    dh = fmaf(-(a_h + b_h), h, a_h);
    dp = (p_inf - p) * inv_tau;
}

// Vectorized path: one float4 per thread per array (global_load_b128 TH_NT).
__global__ __launch_bounds__(256) void hh_vec_kernel(
    const int* __restrict__ tptr,
    const float* __restrict__ Vp, const float* __restrict__ np_,
    const float* __restrict__ mp, const float* __restrict__ hp,
    const float* __restrict__ pp, float* __restrict__ out, int nvec) {
    const int i = blockIdx.x * 256 + threadIdx.x;
    if (i >= nvec) return;

    const float I_in = (tptr[0] > 0) ? HH_I_CONST : 0.0f;   // uniform scalar load

    const v4f V = __builtin_nontemporal_load((const v4f*)Vp  + i);
    const v4f n = __builtin_nontemporal_load((const v4f*)np_ + i);
    const v4f m = __builtin_nontemporal_load((const v4f*)mp  + i);
    const v4f h = __builtin_nontemporal_load((const v4f*)hp  + i);
    const v4f p = __builtin_nontemporal_load((const v4f*)pp  + i);

    v4f dV, dn, dm, dh, dp;
#pragma unroll
    for (int k = 0; k < 4; ++k) {
        float a, b, c, d, e;
        hh_point(V[k], n[k], m[k], h[k], p[k], I_in, a, b, c, d, e);
        dV[k] = a; dn[k] = b; dm[k] = c; dh[k] = d; dp[k] = e;
    }

    v4f* o = (v4f*)out;   // outputs concatenated: dV | dn | dm | dh | dp
    __builtin_nontemporal_store(dV, o + 0 * nvec + i);
    __builtin_nontemporal_store(dn, o + 1 * nvec + i);
    __builtin_nontemporal_store(dm, o + 2 * nvec + i);
    __builtin_nontemporal_store(dh, o + 3 * nvec + i);
    __builtin_nontemporal_store(dp, o + 4 * nvec + i);
}

// Scalar fallback (only used if B is not a multiple of 4).
__global__ __launch_bounds__(256) void hh_scalar_kernel(
    const int* __restrict__ tptr,
    const float* __restrict__ Vp, const float* __restrict__ np_,
    const float* __restrict__ mp, const float* __restrict__ hp,
    const float* __restrict__ pp, float* __restrict__ out, int B) {
    const int i = blockIdx.x * 256 + threadIdx.x;
    if (i >= B) return;
    const float I_in = (tptr[0] > 0) ? HH_I_CONST : 0.0f;
    float a, b, c, d, e;
    hh_point(Vp[i], np_[i], mp[i], hp[i], pp[i], I_in, a, b, c, d, e);
    out[0 * B + i] = a;
    out[1 * B + i] = b;
    out[2 * B + i] = c;
    out[3 * B + i] = d;
    out[4 * B + i] = e;
}

extern "C" void kernel_launch(void* const* d_in, const int* in_sizes, int n_in,
                              void* d_out, int out_size, void* d_ws, size_t ws_size,
                              hipStream_t stream) {
    const int*   t = (const int*)d_in[0];
    const float* V = (const float*)d_in[1];
    const float* n = (const float*)d_in[2];
    const float* m = (const float*)d_in[3];
    const float* h = (const float*)d_in[4];
    const float* p = (const float*)d_in[5];
    float* out = (float*)d_out;
    const int B = in_sizes[1];

    if ((B & 3) == 0) {
        const int nvec   = B >> 2;                 // 2,097,152 float4s for B = 2^23
        const int blocks = (nvec + 255) / 256;     // 8192 blocks of 8 wave32s
        hh_vec_kernel<<<blocks, 256, 0, stream>>>(t, V, n, m, h, p, out, nvec);
    } else {
        const int blocks = (B + 255) / 256;
        hh_scalar_kernel<<<blocks, 256, 0, stream>>>(t, V, n, m, h, p, out, B);
    }
}